// TopKGating_1700807049528
// MI455X (gfx1250) — compile-verified
//
#include <hip/hip_runtime.h>
#include <math.h>

typedef __attribute__((ext_vector_type(2))) float v2f;
typedef __attribute__((ext_vector_type(8))) float v8f;

#define NT 16384
#define DM 2048
#define NE 8
#define TPB 64                 // tokens per block
#define KSPLIT 4               // K-dimension split inside the block
#define KCHUNK (DM / KSPLIT)   // 512 K-elements per wave
#define THREADS 512            // 16 waves: 4 token-groups x 4 K-quarters

// Wave w: token-group (w&3) of 16 tokens, K-quarter (w>>2).
// Each wave runs V_WMMA_F32_16X16X4_F32 over its K range; the 4 partial
// 16x16 accumulators are summed deterministically through LDS.
__global__ __launch_bounds__(THREADS) void moe_gating_kernel(
    const float* __restrict__ x, const float* __restrict__ W,
    float* __restrict__ out, float* __restrict__ ws)
{
  __shared__ float logits_lds[KSPLIT][TPB * 16];  // 16KB partial logits
  __shared__ float red[2 * NE + 1];               // probs_sum[8], freq[8], lse^2

  const int tid = threadIdx.x;
  if (tid < 2 * NE + 1) red[tid] = 0.0f;

  const int wave = tid >> 5;
  const int lane = tid & 31;
  const int tokGroup = wave & 3;
  const int q    = wave >> 2;      // K-quarter
  const int half = lane >> 4;      // 0: K={0,1}, 1: K={2,3} within a chunk
  const int l15  = lane & 15;
  const int tokBase = blockIdx.x * TPB + tokGroup * 16;
  const int kBase   = q * KCHUNK;

  const float* xptr = x + (size_t)(tokBase + l15) * DM + kBase + (half << 1);
  const float* wptr = W + (size_t)(lane & 7) * DM + kBase + (half << 1);

  v8f acc = {};
  for (int k0 = 0; k0 < KCHUNK; k0 += 32) {
    // stream-prefetch ~1KB ahead (speculative: tail overrun is dropped safely)
    __builtin_prefetch(xptr + k0 + 256, 0, 0);
#pragma unroll
    for (int kk = 0; kk < 32; kk += 4) {
      v2f a = *(const v2f*)(xptr + k0 + kk);
      v2f b = *(const v2f*)(wptr + k0 + kk);
      acc = __builtin_amdgcn_wmma_f32_16x16x4_f32(
          /*neg_a=*/false, a, /*neg_b=*/false, b,
          /*c_mod=*/(short)0, acc, /*reuse_a=*/false, /*reuse_b=*/false);
    }
  }

  // D layout: VGPR r, lanes 0-15 -> (token r, expert=lane), lanes 16-31 -> token r+8
  float* dst = &logits_lds[q][(tokGroup * 16 + (half << 3)) * 16 + l15];
#pragma unroll
  for (int r = 0; r < 8; ++r) dst[r * 16] = acc[r];

  __syncthreads();

  if (tid < TPB) {
    const int token = blockIdx.x * TPB + tid;
    float l[NE];
#pragma unroll
    for (int e = 0; e < NE; ++e)
      l[e] = (logits_lds[0][tid * 16 + e] + logits_lds[1][tid * 16 + e]) +
             (logits_lds[2][tid * 16 + e] + logits_lds[3][tid * 16 + e]);

    // top-1 (lowest index wins ties, matching lax.top_k)
    float m1 = l[0]; int i1 = 0;
#pragma unroll
    for (int e = 1; e < NE; ++e) if (l[e] > m1) { m1 = l[e]; i1 = e; }
    // top-2 excluding i1
    float m2 = -__builtin_inff(); int i2 = 0;
#pragma unroll
    for (int e = 0; e < NE; ++e) if (e != i1 && l[e] > m2) { m2 = l[e]; i2 = e; }

    // softmax over the two selected logits
    float t  = expf(m2 - m1);
    float g1 = 1.0f / (1.0f + t);
    float g2 = t / (1.0f + t);

    // dense softmax + logsumexp
    float Z = 0.0f, p[NE];
#pragma unroll
    for (int e = 0; e < NE; ++e) { p[e] = expf(l[e] - m1); Z += p[e]; }
    float invZ = 1.0f / Z;
    float lse  = m1 + logf(Z);

#pragma unroll
    for (int e = 0; e < NE; ++e) atomicAdd(&red[e], p[e] * invZ);
    atomicAdd(&red[NE + i1], 1.0f);
    if (g2 > 0.0f) atomicAdd(&red[NE + i2], 1.0f);
    atomicAdd(&red[2 * NE], lse * lse);

    // outputs: indices (as float), gates, loss slot at 4*NT
    out[token * 2 + 0] = (float)i1;
    out[token * 2 + 1] = (float)i2;
    out[2 * NT + token * 2 + 0] = g1;
    out[2 * NT + token * 2 + 1] = g2;
  }
  __syncthreads();
  if (tid < 2 * NE + 1) atomicAdd(&ws[tid], red[tid]);
}

__global__ void init_ws_kernel(float* __restrict__ ws) {
  int i = threadIdx.x;
  if (i < 2 * NE + 1) ws[i] = 0.0f;
}

__global__ void finalize_loss_kernel(const float* __restrict__ ws,
                                     float* __restrict__ out) {
  if (threadIdx.x == 0 && blockIdx.x == 0) {
    float ps = 0.0f, fs = 0.0f;
    for (int e = 0; e < NE; ++e) { ps += fabsf(ws[e]); fs += fabsf(ws[NE + e]); }
    ps = fmaxf(ps, 1e-12f);
    fs = fmaxf(fs, 1e-12f);
    float s = 0.0f;
    for (int e = 0; e < NE; ++e) s += (ws[e] / ps) * (ws[NE + e] / fs);
    float loss = (float)NE * s + 0.1f * ws[2 * NE] / (float)NT;
    out[4 * NT] = loss;
  }
}

extern "C" void kernel_launch(void* const* d_in, const int* in_sizes, int n_in,
                              void* d_out, int out_size, void* d_ws, size_t ws_size,
                              hipStream_t stream) {
  const float* x = (const float*)d_in[0];   // [16384, 2048] fp32
  const float* W = (const float*)d_in[1];   // [8, 2048] fp32
  float* out = (float*)d_out;               // 65537 floats
  float* ws  = (float*)d_ws;                // 17 floats of scratch

  init_ws_kernel<<<1, 32, 0, stream>>>(ws);
  moe_gating_kernel<<<NT / TPB, THREADS, 0, stream>>>(x, W, out, ws);
  finalize_loss_kernel<<<1, 32, 0, stream>>>(ws, out);
}